// Sketch_89103391523480
// MI455X (gfx1250) — compile-verified
//
#include <hip/hip_runtime.h>

typedef __attribute__((ext_vector_type(2))) float v2f;
typedef __attribute__((ext_vector_type(4))) float f32x4;
typedef __attribute__((ext_vector_type(8))) float v8f;

// ---------------------------------------------------------------------------
// Stage 1: head-mean + 64-token block-mean.
//   in : (4, 32, 8192, 128) fp32
//   out: (4, 128, 128) fp32   (sum over 32 heads * 64 tokens, * 1/2048)
// One workgroup per (tensor, batch, block): streams 1 MiB contiguous-ish.
// ---------------------------------------------------------------------------
__global__ __launch_bounds__(256) void reduce_kernel(
    const float* __restrict__ Q, const float* __restrict__ K,
    float* __restrict__ Qbar, float* __restrict__ Kbar)
{
    const int tid    = threadIdx.x;
    const int dgrp   = tid & 31;   // which float4 of the 128-wide d dim
    const int rowGrp = tid >> 5;   // 8 rows in flight per pass

    const float* __restrict__ src = (blockIdx.y == 0) ? Q : K;
    float* __restrict__       dst = (blockIdx.y == 0) ? Qbar : Kbar;

    const int b   = blockIdx.x >> 7;    // batch 0..3
    const int blk = blockIdx.x & 127;   // token block 0..127

    const size_t headStride = (size_t)8192 * 128;
    const size_t base = (size_t)b * 32 * headStride
                      + (size_t)blk * 64 * 128
                      + (size_t)dgrp * 4;

    f32x4 acc = {0.f, 0.f, 0.f, 0.f};

    // 2048 rows (32 heads x 64 tokens), 8 rows per pass -> 256 passes.
    #pragma unroll 8
    for (int it = 0; it < 256; ++it) {
        const int row = it * 8 + rowGrp;
        const int h   = row >> 6;      // head
        const int t   = row & 63;      // token within block
        const f32x4 v = __builtin_nontemporal_load(
            (const f32x4*)(src + base + (size_t)h * headStride + (size_t)t * 128));
        acc += v;
    }

    __shared__ f32x4 red[256];
    red[tid] = acc;
    __syncthreads();

    if (tid < 32) {
        f32x4 s = red[tid];
        #pragma unroll
        for (int j = 1; j < 8; ++j) s += red[tid + 32 * j];
        s *= (1.0f / 2048.0f);  // mean over 32 heads * 64 tokens
        *(f32x4*)(dst + ((size_t)b * 128 + blk) * 128 + (size_t)tid * 4) = s;
    }
}

// ---------------------------------------------------------------------------
// Stage 2: per batch
//   Qt = (Qbar * rademacher) @ proj   (128x128)@(128x64) -> 128x64
//   Kt = (Kbar * rademacher) @ proj
//   A  = Qt @ Kt^T * 0.125            (128x64)@(64x128)  -> 128x128
// All matmuls via V_WMMA_F32_16X16X4_F32. 8 waves, wave w owns M rows
// [16w, 16w+16). EXEC all ones (no divergence around WMMA).
//
// f32 WMMA fragment layouts (ISA 7.12.2, wave32):
//   A 16x4 : VGPR v, lane l -> M = l%16, K = v + 2*(l>=16)
//   B 4x16 : VGPR v, lane l -> K = v + 2*(l>=16), N = l%16
//   C/D    : VGPR r, lane l -> M = r + 8*(l>=16), N = l%16
// ---------------------------------------------------------------------------
__global__ __launch_bounds__(256) void sketch_kernel(
    const float* __restrict__ Qbar, const float* __restrict__ Kbar,
    const float* __restrict__ rad,  const float* __restrict__ proj,
    float* __restrict__ Ahat)
{
    __shared__ float Qt[128][65];   // pad 64->65: conflict-free column reads
    __shared__ float Kt[128][65];

    const int b    = blockIdx.x;
    const int tid  = threadIdx.x;
    const int w    = tid >> 5;      // wave 0..7
    const int lane = tid & 31;
    const int lm   = lane & 15;
    const int half = lane >> 4;     // 0 or 1

    const float* __restrict__ Qb = Qbar + (size_t)b * 128 * 128;
    const float* __restrict__ Kb = Kbar + (size_t)b * 128 * 128;

    const int m = 16 * w + lm;      // A-fragment row for this lane

    // ---- Qt and Kt: (X * rad) @ proj ------------------------------------
    #pragma unroll
    for (int which = 0; which < 2; ++which) {
        const float* __restrict__ X = which ? Kb : Qb;
        for (int nt = 0; nt < 4; ++nt) {          // N tiles (64 cols)
            const int n = nt * 16 + lm;
            v8f c = {};
            for (int k0 = 0; k0 < 128; k0 += 4) {
                const int ka = k0 + 2 * half;
                v2f a, bf;
                a[0]  = X[m * 128 + ka]     * rad[ka];
                a[1]  = X[m * 128 + ka + 1] * rad[ka + 1];
                bf[0] = proj[ka * 64 + n];
                bf[1] = proj[(ka + 1) * 64 + n];
                c = __builtin_amdgcn_wmma_f32_16x16x4_f32(
                        false, a, false, bf, (short)0, c, false, false);
            }
            #pragma unroll
            for (int r = 0; r < 8; ++r) {
                const int row = 16 * w + r + 8 * half;
                if (which) Kt[row][n] = c[r];
                else       Qt[row][n] = c[r];
            }
        }
    }
    __syncthreads();

    // ---- A = Qt @ Kt^T * 1/sqrt(64) -------------------------------------
    float* __restrict__ out = Ahat + (size_t)b * 128 * 128;
    for (int nt = 0; nt < 8; ++nt) {              // N tiles (128 cols)
        v8f c = {};
        for (int k0 = 0; k0 < 64; k0 += 4) {
            const int ka = k0 + 2 * half;
            v2f a, bf;
            a[0]  = Qt[m][ka];
            a[1]  = Qt[m][ka + 1];
            // B[k][n] = Kt[n][k]  (transposed read, padded stride -> no conflicts)
            bf[0] = Kt[nt * 16 + lm][ka];
            bf[1] = Kt[nt * 16 + lm][ka + 1];
            c = __builtin_amdgcn_wmma_f32_16x16x4_f32(
                    false, a, false, bf, (short)0, c, false, false);
        }
        #pragma unroll
        for (int r = 0; r < 8; ++r) {
            const int row = 16 * w + r + 8 * half;
            out[(size_t)row * 128 + nt * 16 + lm] = 0.125f * c[r];
        }
    }
}

// ---------------------------------------------------------------------------
extern "C" void kernel_launch(void* const* d_in, const int* in_sizes, int n_in,
                              void* d_out, int out_size, void* d_ws, size_t ws_size,
                              hipStream_t stream) {
    (void)in_sizes; (void)n_in; (void)out_size; (void)d_ws; (void)ws_size;

    const float* Q    = (const float*)d_in[0];   // (4,32,8192,128)
    const float* K    = (const float*)d_in[1];   // (4,32,8192,128)
    const float* rad  = (const float*)d_in[2];   // (128,)
    const float* proj = (const float*)d_in[3];   // (128,64)

    float* out  = (float*)d_out;
    const size_t tsz = (size_t)4 * 128 * 128;    // 65536 elements each
    float* Ahat = out;                           // output 0
    float* Qbar = out + tsz;                     // output 1
    float* Kbar = out + 2 * tsz;                 // output 2

    // Stage 1: x = (b*128+blk), y = tensor (0=Q, 1=K)
    reduce_kernel<<<dim3(512, 2), 256, 0, stream>>>(Q, K, Qbar, Kbar);
    // Stage 2: one workgroup per batch
    sketch_kernel<<<4, 256, 0, stream>>>(Qbar, Kbar, rad, proj, Ahat);
}